// LowFER_33054068310429
// MI455X (gfx1250) — compile-verified
//
#include <hip/hip_runtime.h>
#include <math.h>

typedef __attribute__((ext_vector_type(2))) float v2f;
typedef __attribute__((ext_vector_type(8))) float v8f;

#define BATCH   256
#define VOCAB   400000
#define D1      128
#define PP      64
#define KK_     30
#define TT_     20
#define KPAD    32
#define FACTOR  0.040824829046386304f   /* 1/sqrt(30*20) */
#define BN_EPS  1e-5f

// ---------------------------------------------------------------------------
// Kernel 1: front-end. One block per batch element, 64 threads.
// Produces zero-padded x: xp[b][0..29] = bn1(normalized signed-sqrt y), 30..31 = 0
// ---------------------------------------------------------------------------
__global__ __launch_bounds__(64)
void lowfer_front(const int* __restrict__ e1_idx, const int* __restrict__ r_idx,
                  const float* __restrict__ E, const float* __restrict__ R,
                  const float* __restrict__ proj, const int* __restrict__ idx,
                  const float* __restrict__ g0, const float* __restrict__ b0,
                  const float* __restrict__ mu0, const float* __restrict__ vr0,
                  const float* __restrict__ g1, const float* __restrict__ b1,
                  const float* __restrict__ mu1, const float* __restrict__ vr1,
                  float* __restrict__ xp)
{
    __shared__ float e1bn[D1];
    __shared__ float rrow[D1];
    __shared__ float se[PP];
    __shared__ float sr[PP];
    __shared__ float xk[KK_];
    __shared__ float nrm_s;

    const int b = blockIdx.x;
    const int t = threadIdx.x;            // 0..63
    const int e = e1_idx[b];
    const int r = r_idx[b];

    for (int d = t; d < D1; d += 64) {
        float ev = E[(size_t)e * D1 + d];
        e1bn[d]  = (ev - mu0[d]) * rsqrtf(vr0[d] + BN_EPS) * g0[d] + b0[d];
        rrow[d]  = R[(size_t)r * D1 + d];         // D2 == D1 == 128
    }
    __syncthreads();

    float accS = 0.f, accR = 0.f;
    for (int d = 0; d < D1; ++d) {
        float p = proj[d * PP + t];               // coalesced across t
        accS = fmaf(e1bn[d], p, accS);
        accR = fmaf(rrow[d], p, accR);
    }
    se[t] = accS;
    sr[t] = accR;
    __syncthreads();

    if (t < KK_) {
        float y = 0.f;
        for (int u = 0; u < TT_; ++u) {
            int ia = idx[(t * TT_ + u) * 2 + 0];
            int ib = idx[(t * TT_ + u) * 2 + 1];
            y = fmaf(se[ia], sr[ib], y);
        }
        y *= FACTOR;
        float s = (y > 0.f) ? 1.f : ((y < 0.f) ? -1.f : 0.f);
        xk[t] = s * sqrtf(fabsf(y) + 1e-12f);
    }
    __syncthreads();

    if (t == 0) {
        float ss = 0.f;
        for (int k = 0; k < KK_; ++k) ss += xk[k] * xk[k];
        nrm_s = fmaxf(sqrtf(ss), 1e-12f);
    }
    __syncthreads();

    if (t < KPAD) {
        float val = 0.f;
        if (t < KK_) {
            float xx = xk[t] / nrm_s;
            val = (xx - mu1[t]) * rsqrtf(vr1[t] + BN_EPS) * g1[t] + b1[t];
        }
        xp[b * KPAD + t] = val;
    }
}

// ---------------------------------------------------------------------------
// Kernel 2: logits = x @ E[:, :30]^T, sigmoid, store 256 x 400000.
// 256 threads = 8 waves/block; each wave owns a 16-wide v-stripe.
// WMMA f32 16x16x4, K padded to 32 -> 8 chained WMMAs per 16x16 tile.
// Epilogue sigmoid uses v_exp + v_rcp (no precise-divide expansion).
// ---------------------------------------------------------------------------
__global__ __launch_bounds__(256)
void lowfer_gemm(const float* __restrict__ E, const float* __restrict__ xp,
                 float* __restrict__ out)
{
    __shared__ float xs[BATCH * KPAD];            // 32 KB: all of x, zero-padded

    const int tid = threadIdx.x;

    // cooperative load of x into LDS (float4, 2048 vec loads / 256 threads)
    {
        const float4* src = (const float4*)xp;
        float4*       dst = (float4*)xs;
        #pragma unroll
        for (int i = 0; i < (BATCH * KPAD) / (4 * 256); ++i)
            dst[i * 256 + tid] = src[i * 256 + tid];
    }
    __syncthreads();

    const int wave  = tid >> 5;
    const int lane  = tid & 31;
    const int n0    = (blockIdx.x * 8 + wave) * 16;   // this wave's v-stripe
    const int vcol  = n0 + (lane & 15);
    const int khalf = (lane >> 4) * 2;                // 0 (lanes 0-15) or 2 (16-31)

    // B fragments: B[k][n] = E[vcol][k]; VGPR0 = rows k0/k0+2, VGPR1 = k0+1/k0+3.
    // k = 30,31 hold real E data but A is zero there, so products vanish.
    v2f bfrag[8];
    const float* erow = E + (size_t)vcol * D1;
    #pragma unroll
    for (int kk = 0; kk < 8; ++kk) {
        const int kb = kk * 4 + khalf;
        bfrag[kk].x = erow[kb];
        bfrag[kk].y = erow[kb + 1];
    }

    // sweep all 16 batch tiles for this v-stripe
    for (int mt = 0; mt < 16; ++mt) {
        const int m0   = mt * 16;
        const int arow = m0 + (lane & 15);
        v8f acc = {};
        #pragma unroll
        for (int kk = 0; kk < 8; ++kk) {
            const int kb = kk * 4 + khalf;
            v2f a;
            a.x = xs[arow * KPAD + kb];
            a.y = xs[arow * KPAD + kb + 1];
            acc = __builtin_amdgcn_wmma_f32_16x16x4_f32(
                /*neg_a=*/false, a, /*neg_b=*/false, bfrag[kk],
                /*c_mod=*/(short)0, acc, /*reuse_a=*/false, /*reuse_b=*/false);
        }
        // C/D layout: VGPR i -> row m0+i (lanes 0-15) / m0+i+8 (lanes 16-31)
        #pragma unroll
        for (int i = 0; i < 8; ++i) {
            const int m = m0 + i + (lane >> 4) * 8;
            float z  = acc[i];
            // sigmoid via v_exp_f32 + v_rcp_f32 (avoid precise-divide expansion)
            float sg = __builtin_amdgcn_rcpf(1.0f + __expf(-z));
            out[(size_t)m * VOCAB + vcol] = sg;
        }
    }
}

// ---------------------------------------------------------------------------
extern "C" void kernel_launch(void* const* d_in, const int* in_sizes, int n_in,
                              void* d_out, int out_size, void* d_ws, size_t ws_size,
                              hipStream_t stream)
{
    const int*   e1_idx = (const int*)  d_in[0];
    const int*   r_idx  = (const int*)  d_in[1];
    const float* E      = (const float*)d_in[2];
    const float* R      = (const float*)d_in[3];
    const float* proj   = (const float*)d_in[4];
    const int*   idx    = (const int*)  d_in[5];
    const float* g0     = (const float*)d_in[6];
    const float* b0     = (const float*)d_in[7];
    const float* mu0    = (const float*)d_in[8];
    const float* vr0    = (const float*)d_in[9];
    const float* g1     = (const float*)d_in[10];
    const float* b1     = (const float*)d_in[11];
    const float* mu1    = (const float*)d_in[12];
    const float* vr1    = (const float*)d_in[13];
    float* out = (float*)d_out;
    float* xp  = (float*)d_ws;            // 256*32*4 = 32 KB scratch

    lowfer_front<<<BATCH, 64, 0, stream>>>(e1_idx, r_idx, E, R, proj, idx,
                                           g0, b0, mu0, vr0, g1, b1, mu1, vr1, xp);

    lowfer_gemm<<<VOCAB / 128, 256, 0, stream>>>(E, xp, out);
}